// SMPL_10153302687829
// MI455X (gfx1250) — compile-verified
//
#include <hip/hip_runtime.h>

typedef __attribute__((ext_vector_type(2))) float v2f;
typedef __attribute__((ext_vector_type(8))) float v8f;

#define N_J 24
#define N_B 10
#define N_P 207        // (24-1)*9
#define BETA_STRIDE 12 // 10 padded to 12, zeros at [10],[11]
#define LROT_STRIDE 208
#define G_STRIDE 288   // 24*12

__device__ __constant__ int c_parents[N_J] = {0, 0, 0, 0, 1, 2, 3, 4, 5, 6, 7, 8,
                                              9, 9, 9, 12, 13, 14, 16, 17, 18, 19, 20, 21};

// D = A(16x4 f32) * B(4x16 f32) + C(16x16 f32)
__device__ inline v8f wmma4(v2f a, v2f b, v8f c) {
    return __builtin_amdgcn_wmma_f32_16x16x4_f32(
        /*neg_a=*/false, a, /*neg_b=*/false, b,
        /*c_mod=*/(short)0, c, /*reuse_a=*/false, /*reuse_b=*/false);
}

// ---------------------------------------------------------------------------
// Kernel 1: JT[24*3] = Jreg @ v_template ; JS[24*3*10] = Jreg @ shapedirs
// One wave per output element; 792 = 24*(3 + 30) blocks.
// ---------------------------------------------------------------------------
__global__ __launch_bounds__(32) void smpl_jreg(const float* __restrict__ Jr,
                                                const float* __restrict__ vt,
                                                const float* __restrict__ sd,
                                                float* __restrict__ JT,
                                                float* __restrict__ JS, int V) {
    const int o = blockIdx.x;
    const int lane = threadIdx.x;
    float acc = 0.0f;
    if (o < 72) {
        const int j = o / 3, c = o % 3;
        for (int v = lane; v < V; v += 32) acc += Jr[j * V + v] * vt[v * 3 + c];
    } else {
        const int q = o - 72;
        const int j = q / 30, r = q % 30;   // r = c*10 + b
        for (int v = lane; v < V; v += 32) acc += Jr[j * V + v] * sd[v * 30 + r];
    }
    for (int off = 16; off > 0; off >>= 1) acc += __shfl_down(acc, off, 32);
    if (lane == 0) {
        if (o < 72) JT[o] = acc; else JS[o - 72] = acc;
    }
}

// ---------------------------------------------------------------------------
// Kernel 2: per-sample Rodrigues, lrotmin, padded beta, joint locations,
// kinematic chain, rest-pose fixup.  One thread per sample n.
//   beta_pad[n*12 + b]          (b<10 copy; 10,11 zero)
//   lrot[n*208 + p]             (p<207; element 207 zeroed)
//   G  [n*288 + j*12 + a*4 + b] (a in 0..2, b in 0..3)
// ---------------------------------------------------------------------------
__global__ void smpl_prep(const float* __restrict__ beta,
                          const float* __restrict__ pose,
                          const float* __restrict__ JT,
                          const float* __restrict__ JS,
                          float* __restrict__ beta_pad,
                          float* __restrict__ lrot,
                          float* __restrict__ G, int N) {
    const int n = blockIdx.x * blockDim.x + threadIdx.x;
    if (n >= N) return;

    float Rm[N_J][9];
    float Jl[N_J][3];
    float Gl[N_J][12];

    // padded beta
    for (int b = 0; b < N_B; ++b) beta_pad[n * BETA_STRIDE + b] = beta[n * N_B + b];
    beta_pad[n * BETA_STRIDE + 10] = 0.0f;
    beta_pad[n * BETA_STRIDE + 11] = 0.0f;

    // Rodrigues per joint
    for (int j = 0; j < N_J; ++j) {
        const float rx = pose[n * (N_J * 3) + j * 3 + 0];
        const float ry = pose[n * (N_J * 3) + j * 3 + 1];
        const float rz = pose[n * (N_J * 3) + j * 3 + 2];
        const float th = sqrtf(rx * rx + ry * ry + rz * rz) + 1e-8f;
        const float inv = 1.0f / th;
        const float ux = rx * inv, uy = ry * inv, uz = rz * inv;
        const float cc = cosf(th), ss = sinf(th), tt = 1.0f - cc;
        Rm[j][0] = cc + tt * ux * ux;      Rm[j][1] = tt * ux * uy - ss * uz;  Rm[j][2] = tt * ux * uz + ss * uy;
        Rm[j][3] = tt * ux * uy + ss * uz; Rm[j][4] = cc + tt * uy * uy;       Rm[j][5] = tt * uy * uz - ss * ux;
        Rm[j][6] = tt * ux * uz - ss * uy; Rm[j][7] = tt * uy * uz + ss * ux;  Rm[j][8] = cc + tt * uz * uz;
    }

    // lrotmin = (R[1:] - I).flatten  (padded stride 208)
    for (int j = 1; j < N_J; ++j)
        for (int k = 0; k < 9; ++k) {
            const float eye = (k == 0 || k == 4 || k == 8) ? 1.0f : 0.0f;
            lrot[(size_t)n * LROT_STRIDE + (j - 1) * 9 + k] = Rm[j][k] - eye;
        }
    lrot[(size_t)n * LROT_STRIDE + N_P] = 0.0f;

    // joint locations: J = JT + JS @ beta[n]
    for (int j = 0; j < N_J; ++j)
        for (int c = 0; c < 3; ++c) {
            float acc = JT[j * 3 + c];
            for (int b = 0; b < N_B; ++b)
                acc += JS[j * 30 + c * 10 + b] * beta[n * N_B + b];
            Jl[j][c] = acc;
        }

    // kinematic chain (3x4 transforms; implicit bottom row [0,0,0,1])
    for (int a = 0; a < 3; ++a) {
        for (int b = 0; b < 3; ++b) Gl[0][a * 4 + b] = Rm[0][a * 3 + b];
        Gl[0][a * 4 + 3] = Jl[0][a];
    }
    for (int i = 1; i < N_J; ++i) {
        const int p = c_parents[i];
        const float t0 = Jl[i][0] - Jl[p][0];
        const float t1 = Jl[i][1] - Jl[p][1];
        const float t2 = Jl[i][2] - Jl[p][2];
        for (int a = 0; a < 3; ++a) {
            const float g0 = Gl[p][a * 4 + 0], g1 = Gl[p][a * 4 + 1], g2 = Gl[p][a * 4 + 2];
            for (int b = 0; b < 3; ++b)
                Gl[i][a * 4 + b] = g0 * Rm[i][b] + g1 * Rm[i][3 + b] + g2 * Rm[i][6 + b];
            Gl[i][a * 4 + 3] = g0 * t0 + g1 * t1 + g2 * t2 + Gl[p][a * 4 + 3];
        }
    }

    // remove rest-pose joint translation and store
    for (int i = 0; i < N_J; ++i)
        for (int a = 0; a < 3; ++a) {
            const float g0 = Gl[i][a * 4 + 0], g1 = Gl[i][a * 4 + 1], g2 = Gl[i][a * 4 + 2];
            const float tr = Gl[i][a * 4 + 3] - (g0 * Jl[i][0] + g1 * Jl[i][1] + g2 * Jl[i][2]);
            float* dst = G + (size_t)n * G_STRIDE + i * 12 + a * 4;
            dst[0] = g0; dst[1] = g1; dst[2] = g2; dst[3] = tr;
        }
}

// ---------------------------------------------------------------------------
// Kernel 3: fused blendshapes + LBS.  One wave per 16v x 16n tile.
// A-fragment (16x4 f32): lane l<16 -> M=l, K={0,1}; lane 16+l -> M=l, K={2,3}
// B-fragment (4x16 f32): lane l<16 -> N=l, K={0,1}; lane 16+l -> N=l, K={2,3}
// C/D: VGPR r, lane l<16 -> (M=r, N=l); lane 16+l -> (M=r+8, N=l)
// All fragment loads are UNCONDITIONAL: B side is zero-padded in memory, A
// side uses index clamps (garbage * 0 = 0), so no exec-mask churn near WMMA.
// ---------------------------------------------------------------------------
__global__ __launch_bounds__(32) void smpl_lbs(const float* __restrict__ beta_pad,
                                               const float* __restrict__ vtpl,
                                               const float* __restrict__ sdirs,
                                               const float* __restrict__ pdirs,
                                               const float* __restrict__ wts,
                                               const float* __restrict__ lrot,
                                               const float* __restrict__ G,
                                               float* __restrict__ out,
                                               int N, int V) {
    const int lane = threadIdx.x;
    const int half = lane >> 4;       // 0 or 1
    const int l    = lane & 15;
    const int vt = blockIdx.x * 16;
    const int nt = blockIdx.y * 16;

    const int va = (vt + l < V) ? (vt + l) : (V - 1);   // clamped A-row vertex
    const int nb = (nt + l < N) ? (nt + l) : (N - 1);   // clamped B-col sample
    const int k2 = half * 2;                            // lane K offset

    // per-lane base pointers (computed once)
    const float* pBeta = beta_pad + nb * BETA_STRIDE + k2;     // zeros at 10,11
    const float* pLrot = lrot + (size_t)nb * LROT_STRIDE + k2; // zero at 207
    const float* pWts  = wts + va * N_J + k2;
    const float* pG    = G + (size_t)nb * G_STRIDE;

    v8f P3[3];
#pragma unroll
    for (int c = 0; c < 3; ++c) {
        const float* pAs = sdirs + va * (3 * N_B) + c * N_B;  // shapedirs row
        const float* pAp = pdirs + va * (3 * N_P) + c * N_P;  // posedirs row

        v8f acc = {};
        // ---- shapedirs: K = 10 padded to 12 ----
#pragma unroll
        for (int kk = 0; kk < 2; ++kk) {           // K 0..7: in range for all lanes
            v2f a, b;
            a.x = pAs[kk * 4 + k2];
            a.y = pAs[kk * 4 + k2 + 1];
            b.x = pBeta[kk * 4];
            b.y = pBeta[kk * 4 + 1];
            acc = wmma4(a, b, acc);
        }
        {   // K 8..11: clamp A to <=9; B is exactly zero at 10,11
            int i0 = 8 + k2;     i0 = (i0 < N_B) ? i0 : (N_B - 1);
            int i1 = 8 + k2 + 1; i1 = (i1 < N_B) ? i1 : (N_B - 1);
            v2f a, b;
            a.x = pAs[i0];
            a.y = pAs[i1];
            b.x = pBeta[8];
            b.y = pBeta[9];
            acc = wmma4(a, b, acc);
        }
        // ---- posedirs: K = 207 padded to 208 ----
#pragma unroll
        for (int kk = 0; kk < 51; ++kk) {          // K 0..203: in range for all lanes
            v2f a, b;
            a.x = pAp[kk * 4 + k2];
            a.y = pAp[kk * 4 + k2 + 1];
            b.x = pLrot[kk * 4];
            b.y = pLrot[kk * 4 + 1];
            acc = wmma4(a, b, acc);
        }
        {   // K 204..207: clamp A to <=206; B is exactly zero at 207
            int i0 = 204 + k2;     i0 = (i0 < N_P) ? i0 : (N_P - 1);
            int i1 = 204 + k2 + 1; i1 = (i1 < N_P) ? i1 : (N_P - 1);
            v2f a, b;
            a.x = pAp[i0];
            a.y = pAp[i1];
            b.x = pLrot[204];
            b.y = pLrot[205];
            acc = wmma4(a, b, acc);
        }
        // ---- + v_template (clamped unconditional loads) ----
#pragma unroll
        for (int r = 0; r < 8; ++r) {
            int v = vt + r + half * 8;
            v = (v < V) ? v : (V - 1);
            acc[r] += vtpl[v * 3 + c];
        }
        P3[c] = acc;
    }

    // ---- skinning: preload weight A-fragments (K = 24 = 6*4, no padding) ----
    v2f wA[6];
#pragma unroll
    for (int kk = 0; kk < 6; ++kk) {
        wA[kk].x = pWts[kk * 4];
        wA[kk].y = pWts[kk * 4 + 1];
    }

    const bool interior = (vt + 16 <= V) && (nt + 16 <= N);   // block-uniform

#pragma unroll
    for (int a = 0; a < 3; ++a) {
        v8f T[4];
#pragma unroll
        for (int b = 0; b < 4; ++b) {
            v8f t = {};
#pragma unroll
            for (int kk = 0; kk < 6; ++kk) {
                const int j0 = kk * 4 + k2;
                v2f bb;
                bb.x = pG[j0 * 12 + a * 4 + b];
                bb.y = pG[(j0 + 1) * 12 + a * 4 + b];
                t = wmma4(wA[kk], bb, t);
            }
            T[b] = t;
        }
        if (interior) {
#pragma unroll
            for (int r = 0; r < 8; ++r) {
                const float val = T[0][r] * P3[0][r] + T[1][r] * P3[1][r] +
                                  T[2][r] * P3[2][r] + T[3][r];
                const int v = vt + r + half * 8;
                out[(size_t)(nt + l) * V * 3 + v * 3 + a] = val;
            }
        } else {
#pragma unroll
            for (int r = 0; r < 8; ++r) {
                const float val = T[0][r] * P3[0][r] + T[1][r] * P3[1][r] +
                                  T[2][r] * P3[2][r] + T[3][r];
                const int v = vt + r + half * 8;
                const int n = nt + l;
                if (v < V && n < N)
                    out[(size_t)n * V * 3 + v * 3 + a] = val;
            }
        }
    }
}

// ---------------------------------------------------------------------------
extern "C" void kernel_launch(void* const* d_in, const int* in_sizes, int n_in,
                              void* d_out, int out_size, void* d_ws, size_t ws_size,
                              hipStream_t stream) {
    const float* beta  = (const float*)d_in[0];
    const float* pose  = (const float*)d_in[1];
    const float* vtpl  = (const float*)d_in[2];
    const float* sdirs = (const float*)d_in[3];
    const float* pdirs = (const float*)d_in[4];
    const float* Jreg  = (const float*)d_in[5];
    const float* wts   = (const float*)d_in[6];

    const int N = in_sizes[0] / N_B;      // 512
    const int V = in_sizes[2] / 3;        // 6890

    float* ws     = (float*)d_ws;
    float* wsJT   = ws;                                   // 72
    float* wsJS   = ws + 72;                              // 720
    float* wsBeta = ws + 800;                             // N*12
    float* wsLrot = wsBeta + (size_t)N * BETA_STRIDE;     // N*208
    float* wsG    = wsLrot + (size_t)N * LROT_STRIDE;     // N*288

    smpl_jreg<<<24 * 33, 32, 0, stream>>>(Jreg, vtpl, sdirs, wsJT, wsJS, V);
    smpl_prep<<<(N + 63) / 64, 64, 0, stream>>>(beta, pose, wsJT, wsJS,
                                                wsBeta, wsLrot, wsG, N);

    dim3 grid((V + 15) / 16, (N + 15) / 16);
    smpl_lbs<<<grid, 32, 0, stream>>>(wsBeta, vtpl, sdirs, pdirs, wts, wsLrot, wsG,
                                      (float*)d_out, N, V);
}